// VectorQuantiser_28656021799185
// MI455X (gfx1250) — compile-verified
//
#include <hip/hip_runtime.h>

// ---------------------------------------------------------------------------
// VQ codebook search for MI455X (gfx1250, wave32).
//   z: (8,4096,64) f32, codebook: (8192,64) f32
//   out: z_q (B*T*64 f32) ++ loss (1 f32)
//
// Cross-term GEMM via V_WMMA_F32_16X16X4_F32 (full fp32: codebook entries are
// ~1e-4, argmin gaps ~1e-6 -- f16/bf16 mantissas can't resolve them).
// Each wave keeps TWO 16x64 z tiles (32 rows) in registers and reuses every
// codebook B-fragment for two WMMAs: 16 loads : 32 wmma per 16-code tile,
// halving VMEM traffic per output row vs the 1:1 version.
// ---------------------------------------------------------------------------

typedef float v2f __attribute__((ext_vector_type(2)));
typedef float v8f __attribute__((ext_vector_type(8)));

#define BT            32768      // B*T
#define DIM           64         // D
#define KCB           8192       // K
#define ROWS_PER_WAVE 32         // two 16-row WMMA C tiles per wave
#define WAVES_PER_WG  4
#define ROWS_PER_WG   (ROWS_PER_WAVE * WAVES_PER_WG)   // 128
#define NUM_WG        (BT / ROWS_PER_WG)               // 256
#define NTILES        (KCB / 16)                       // 512
#define NPART         (NUM_WG * WAVES_PER_WG)          // 1024 loss partials

// ---- prep: e2[k] = ||codebook[k]||^2 --------------------------------------
__global__ __launch_bounds__(256)
void vq_prep(const float* __restrict__ cb, float* __restrict__ e2) {
  int k = blockIdx.x * blockDim.x + threadIdx.x;
  if (k < KCB) {
    const float4* p = (const float4*)(cb + (size_t)k * DIM);
    float s = 0.f;
#pragma unroll
    for (int i = 0; i < DIM / 4; ++i) {
      float4 v = p[i];
      s = __builtin_fmaf(v.x, v.x, s);
      s = __builtin_fmaf(v.y, v.y, s);
      s = __builtin_fmaf(v.z, v.z, s);
      s = __builtin_fmaf(v.w, v.w, s);
    }
    e2[k] = s;
  }
}

// ---- main: fused WMMA cross-term + argmin + gather + loss partials --------
__global__ __launch_bounds__(128)
void vq_main(const float* __restrict__ z, const float* __restrict__ cb,
             const float* __restrict__ e2, float* __restrict__ out,
             float* __restrict__ partials) {
  const int lane   = threadIdx.x & 31;
  const int wave   = threadIdx.x >> 5;
  const int half   = lane >> 4;     // 0: lanes 0-15, 1: lanes 16-31
  const int lane16 = lane & 15;
  const int row0   = blockIdx.x * ROWS_PER_WG + wave * ROWS_PER_WAVE;

  // Two A tiles: rows [row0, row0+16) and [row0+16, row0+32), 64 VGPRs total.
  // f32 16x4 A layout: VGPR j, lane-half h -> k_local = j + 2*h, so each lane
  // loads the consecutive dim pair {dd*4 + 2*half, +1} as one v2f.
  v2f a0[16], a1[16];
  {
    const float* zp0 = z + (size_t)(row0 + lane16) * DIM + 2 * half;
    const float* zp1 = zp0 + 16 * DIM;
#pragma unroll
    for (int dd = 0; dd < 16; ++dd) {
      a0[dd] = *(const v2f*)(zp0 + dd * 4);
      a1[dd] = *(const v2f*)(zp1 + dd * 4);
    }
  }

  // Running per-lane argmin of (e2[k] - 2*cross). Lane covers code N = lane16.
  //   entry v in [0,8):  row = row0 + v + 8*half          (C tile 0)
  //   entry v in [8,16): row = row0 + 16 + (v-8) + 8*half (C tile 1)
  float minv[16];
  int   mini[16];
#pragma unroll
  for (int v = 0; v < 16; ++v) { minv[v] = 3.0e38f; mini[v] = 0; }

  const float* cbl = cb + (size_t)lane16 * DIM + 2 * half;

  for (int kt = 0; kt < NTILES; ++kt) {
    const float* bp = cbl + (size_t)kt * 16 * DIM;
    __builtin_prefetch(bp + 16 * DIM, 0, 0);   // next tile -> global_prefetch_b8
    float e2lane = e2[kt * 16 + lane16];

    v8f c0 = {}, c1 = {};
#pragma unroll
    for (int dd = 0; dd < 16; ++dd) {
      // B 4x16 f32 layout mirrors A: VGPR j, half h -> K = j + 2*h; N = lane16.
      v2f b = *(const v2f*)(bp + dd * 4);
      c0 = __builtin_amdgcn_wmma_f32_16x16x4_f32(
               false, a0[dd], false, b, (short)0, c0, false, false);
      c1 = __builtin_amdgcn_wmma_f32_16x16x4_f32(
               false, a1[dd], false, b, (short)0, c1, false, false);
    }

    const int kidx = kt * 16 + lane16;
#pragma unroll
    for (int v = 0; v < 8; ++v) {
      float s0 = __builtin_fmaf(-2.0f, c0[v], e2lane);  // z^2 is row-constant
      float s1 = __builtin_fmaf(-2.0f, c1[v], e2lane);
      if (s0 < minv[v])     { minv[v]     = s0; mini[v]     = kidx; }
      if (s1 < minv[v + 8]) { minv[v + 8] = s1; mini[v + 8] = kidx; }
    }
  }

  // Butterfly argmin across the 16 lanes of each half (wave32 shuffles).
  // Tie-break on smaller index to match jnp.argmin (first occurrence).
#pragma unroll
  for (int off = 1; off < 16; off <<= 1) {
#pragma unroll
    for (int v = 0; v < 16; ++v) {
      float ov = __shfl_xor(minv[v], off, 32);
      int   oi = __shfl_xor(mini[v], off, 32);
      if (ov < minv[v] || (ov == minv[v] && oi < mini[v])) {
        minv[v] = ov; mini[v] = oi;
      }
    }
  }

  // Gather winning codebook rows, write z_q, accumulate (q - z)^2.
  float lossAcc = 0.f;
#pragma unroll
  for (int m = 0; m < ROWS_PER_WAVE; ++m) {
    const int tile = m >> 4;            // which C tile
    const int mt   = m & 15;            // row within tile
    const int v    = (mt & 7) + 8 * tile;
    const int src  = (mt >> 3) << 4;    // lane half holding this row
    int idx = __shfl(mini[v], src, 32);

    const float* qrow = cb + (size_t)idx * DIM;
    const int    row  = row0 + m;
    const float* zr   = z   + (size_t)row * DIM;
    float*       orow = out + (size_t)row * DIM;

    v2f q  = *(const v2f*)(qrow + lane * 2);
    v2f zz = *(const v2f*)(zr   + lane * 2);
    *(v2f*)(orow + lane * 2) = q;                      // z_q_ste == z_q numerically

    float d0 = q.x - zz.x, d1 = q.y - zz.y;
    lossAcc = __builtin_fmaf(d0, d0, lossAcc);
    lossAcc = __builtin_fmaf(d1, d1, lossAcc);
  }

  // Wave-reduce loss, one partial per wave (deterministic final sum).
#pragma unroll
  for (int off = 1; off < 32; off <<= 1) lossAcc += __shfl_xor(lossAcc, off, 32);
  if (lane == 0) partials[blockIdx.x * WAVES_PER_WG + wave] = lossAcc;
}

// ---- finish: fixed-order sum of partials -> loss scalar -------------------
__global__ void vq_finish(const float* __restrict__ partials,
                          float* __restrict__ out_loss) {
  float s = 0.f;
  for (int i = 0; i < NPART; ++i) s += partials[i];
  *out_loss = s * (1.25f / (float)(BT * DIM));   // 0.25*mean + 1.0*mean
}

// ---------------------------------------------------------------------------
extern "C" void kernel_launch(void* const* d_in, const int* in_sizes, int n_in,
                              void* d_out, int out_size, void* d_ws, size_t ws_size,
                              hipStream_t stream) {
  (void)in_sizes; (void)n_in; (void)out_size; (void)ws_size;
  const float* z  = (const float*)d_in[0];   // (8,4096,64) f32
  const float* cb = (const float*)d_in[1];   // (8192,64)   f32
  float* out = (float*)d_out;                // z_q ++ loss

  float* e2       = (float*)d_ws;            // KCB floats
  float* partials = e2 + KCB;                // NPART floats

  vq_prep  <<<KCB / 256, 256, 0, stream>>>(cb, e2);
  vq_main  <<<NUM_WG,    128, 0, stream>>>(z, cb, e2, out, partials);
  vq_finish<<<1,         1,   0, stream>>>(partials, out + (size_t)BT * DIM);
}